// Model_82789789598332
// MI455X (gfx1250) — compile-verified
//
#include <hip/hip_runtime.h>
#include <hip/hip_bf16.h>

#define IMG_DIM     768
#define PIX_PER_BLK 256
#define BLKS_PER_ROW (IMG_DIM / PIX_PER_BLK)   // 3
#define NCH         17
#define NOFF        34

typedef int   v4i_t __attribute__((ext_vector_type(4)));
typedef float v4f_t __attribute__((ext_vector_type(4)));
typedef v4i_t __attribute__((address_space(1)))* gptr_v4i;   // global
typedef v4i_t __attribute__((address_space(3)))* lptr_v4i;   // LDS

// ---- CDNA5 async global->LDS copy (ASYNCcnt path) -------------------------
__device__ __forceinline__ void async_copy_f4(void* lds_dst, const void* gsrc) {
#if __has_builtin(__builtin_amdgcn_global_load_async_to_lds_b128)
    __builtin_amdgcn_global_load_async_to_lds_b128(
        (gptr_v4i)gsrc,          // global src, AS1, v4i*
        (lptr_v4i)lds_dst,       // LDS dst,   AS3, v4i*
        /*imm offset=*/0, /*cpol=*/0);
#else
    uint32_t lds_addr =
        (uint32_t)(uintptr_t)(__attribute__((address_space(3))) void*)lds_dst;
    uint64_t gaddr = (uint64_t)(uintptr_t)gsrc;
    asm volatile("global_load_async_to_lds_b128 %0, %1, off"
                 :: "v"(lds_addr), "v"(gaddr)
                 : "memory");
#endif
}

__device__ __forceinline__ void wait_async0() {
#if __has_builtin(__builtin_amdgcn_s_wait_asynccnt)
    __builtin_amdgcn_s_wait_asynccnt(0);
#else
    asm volatile("s_wait_asynccnt 0" ::: "memory");
#endif
}

// ---- kernel ---------------------------------------------------------------
__global__ __launch_bounds__(PIX_PER_BLK) void pose_decode_kernel(
    const float* __restrict__ heat,   // (768*768, 17) contiguous per pixel
    const float* __restrict__ off,    // (768*768, 34): [0..16]=dy, [17..33]=dx
    float* __restrict__ out)          // (768*768, 4): [cls, score, x, y]
{
    __shared__ float ldsH[PIX_PER_BLK * NCH];   // 17408 B
    __shared__ float ldsO[PIX_PER_BLK * NOFF];  // 34816 B

    const int tid      = threadIdx.x;
    const int blk      = blockIdx.x;
    const int base_pix = blk * PIX_PER_BLK;

    // Stage both tiles with coalesced, 16B-aligned async b128 loads.
    // Straight-line issue: fixed trip counts, no exec-masked loop overhead.
    const v4f_t* gH = (const v4f_t*)(heat + (size_t)base_pix * NCH);
    const v4f_t* gO = (const v4f_t*)(off  + (size_t)base_pix * NOFF);
    v4f_t* lH = (v4f_t*)ldsH;
    v4f_t* lO = (v4f_t*)ldsO;

    // heat tile: 256*17/4 = 1088 chunks = 4*256 + 64
#pragma unroll
    for (int r = 0; r < 4; ++r)
        async_copy_f4(lH + r * PIX_PER_BLK + tid, gH + r * PIX_PER_BLK + tid);
    if (tid < 64)
        async_copy_f4(lH + 4 * PIX_PER_BLK + tid, gH + 4 * PIX_PER_BLK + tid);

    // offsets tile: 256*34/4 = 2176 chunks = 8*256 + 128
#pragma unroll
    for (int r = 0; r < 8; ++r)
        async_copy_f4(lO + r * PIX_PER_BLK + tid, gO + r * PIX_PER_BLK + tid);
    if (tid < 128)
        async_copy_f4(lO + 8 * PIX_PER_BLK + tid, gO + 8 * PIX_PER_BLK + tid);

    wait_async0();       // this wave's async transfers landed in LDS
    __syncthreads();     // publish across the workgroup

    // Argmax over 17 channels (first-max semantics, matching jnp.argmax).
    // LDS dword stride 17 across lanes -> conflict-free (17 coprime with 64).
    const float* hrow = ldsH + tid * NCH;
    float best = hrow[0];
    int   bidx = 0;
#pragma unroll
    for (int c = 1; c < NCH; ++c) {
        float v = hrow[c];
        if (v > best) { best = v; bidx = c; }
    }

    // Row = exactly 3 blocks: scalar h/w derivation (SALU), no per-lane div.
    const int h = blk / BLKS_PER_ROW;
    const int w = (blk - h * BLKS_PER_ROW) * PIX_PER_BLK + tid;
    const int p = base_pix + tid;

    const float* orow = ldsO + tid * NOFF;      // stride 34 -> conflict-free
    float offy = orow[bidx];
    float offx = orow[NCH + bidx];

    // trunc-toward-zero matches astype(int32)
    float x = (float)(int)((float)(w << 2) + offx);
    float y = (float)(int)((float)(h << 2) + offy);
    float score = 1.0f / (1.0f + __expf(-best));

    v4f_t r;
    r.x = (float)bidx;  // classid
    r.y = score;
    r.z = x;
    r.w = y;
    __builtin_nontemporal_store(r, (v4f_t*)out + p);   // coalesced 16B NT store
}

// ---- launch ---------------------------------------------------------------
extern "C" void kernel_launch(void* const* d_in, const int* in_sizes, int n_in,
                              void* d_out, int out_size, void* d_ws, size_t ws_size,
                              hipStream_t stream) {
    const float* heat = (const float*)d_in[0];  // (1,768,768,17) f32
    const float* off  = (const float*)d_in[1];  // (1,768,768,34) f32
    float* out        = (float*)d_out;          // (1, 589824, 4) f32

    const int npix  = IMG_DIM * IMG_DIM;        // 589824, divisible by 256
    const int nblks = npix / PIX_PER_BLK;       // 2304

    pose_decode_kernel<<<nblks, PIX_PER_BLK, 0, stream>>>(heat, off, out);
}